// WanS2VTransformer3DModel_19181323944225
// MI455X (gfx1250) — compile-verified
//
#include <hip/hip_runtime.h>
#include <stdint.h>

// ---------------------------------------------------------------------------
// 3D RoPE (WanS2V) for MI455X / gfx1250.
//
// Bandwidth-bound: 268MB read + 268MB write; floor ~23us at 23.3 TB/s.
// Strategy:
//   * one 256-thread block processes 16 tiles of 256 float4 (64KB in/out)
//   * CDNA5 async global->LDS (b128) double-buffered pipeline, per-wave
//     s_wait_asynccnt, no barriers (each lane reads back only its own 16B)
//   * trig gather (tiny, L2-resident) recomputed only when position changes
//   * non-temporal b128 stores (and NT fallback loads) to avoid L2 thrash
// ---------------------------------------------------------------------------

#define TPB              256
#define TILES_PER_BLOCK  16
#define F_DIM            22
#define H_DIM            21
#define NUM_CH           64     // d/2
// float4s per position = n*d/4 = 16*128/4 = 512  -> shift by 9

typedef float v4f __attribute__((ext_vector_type(4)));
typedef int   v4i __attribute__((ext_vector_type(4)));

// Address-space-qualified pointer types matching the builtin's signature
// (clang diagnostic showed param 0 is "int4 __device__ *", i.e. AS1 int4*).
typedef __attribute__((address_space(1))) v4i* glob_b128_t;
typedef __attribute__((address_space(3))) v4i* lds_b128_t;

#if defined(__HIP_DEVICE_COMPILE__) && \
    __has_builtin(__builtin_amdgcn_global_load_async_to_lds_b128)
#define HAVE_ASYNC_LDS 1
#else
#define HAVE_ASYNC_LDS 0
#endif

#if defined(__HIP_DEVICE_COMPILE__) && !HAVE_ASYNC_LDS
#warning "gfx1250: __builtin_amdgcn_global_load_async_to_lds_b128 NOT available; using direct global loads"
#endif

#if HAVE_ASYNC_LDS
__device__ __forceinline__ void async_copy_b128(v4f* lds_dst, const v4f* g_src) {
  __builtin_amdgcn_global_load_async_to_lds_b128(
      (glob_b128_t)g_src,
      (lds_b128_t)lds_dst,
      /*offset=*/0, /*cpol=*/0);
}

template <int N>
__device__ __forceinline__ void wait_async() {
#if __has_builtin(__builtin_amdgcn_s_wait_asynccnt)
  __builtin_amdgcn_s_wait_asynccnt(N);
#else
  asm volatile("s_wait_asynccnt %0" ::"n"(N) : "memory");
#endif
}
#endif  // HAVE_ASYNC_LDS

__global__ __launch_bounds__(TPB) void rope3d_wan_s2v_kernel(
    const float* __restrict__ x,
    const float* __restrict__ cos_tab,
    const float* __restrict__ sin_tab,
    const int* __restrict__ h_ptr,
    const int* __restrict__ w_ptr,
    float* __restrict__ out,
    int n_f4) {
  const int tid = threadIdx.x;
  const int h = *h_ptr;   // scalar loads (uniform)
  const int w = *w_ptr;

  // This thread's channel pair is invariant: pairs (2k, 2k+1), k = tid & 31.
  const int c0 = (tid & 31) * 2;
  const int c1 = c0 + 1;

  const int blockBase  = (int)blockIdx.x * (TILES_PER_BLOCK * TPB);
  const bool blockFull = (blockBase + TILES_PER_BLOCK * TPB) <= n_f4;

  const v4f* __restrict__ x4   = (const v4f*)x;
  v4f* __restrict__       out4 = (v4f*)out;

  float gc0 = 0.f, gs0 = 0.f, gc1 = 0.f, gs1 = 0.f;
  int cur_s = -1;

  auto load_trig = [&](int s) {
    int t1 = s / w;
    int wi = s - t1 * w;
    int fi = t1 / h;
    int hi = t1 - fi * h;
    int r0 = (c0 < F_DIM) ? fi : ((c0 < F_DIM + H_DIM) ? hi : wi);
    int r1 = (c1 < F_DIM) ? fi : ((c1 < F_DIM + H_DIM) ? hi : wi);
    gc0 = cos_tab[r0 * NUM_CH + c0];
    gs0 = sin_tab[r0 * NUM_CH + c0];
    gc1 = cos_tab[r1 * NUM_CH + c1];
    gs1 = sin_tab[r1 * NUM_CH + c1];
  };

  auto rotate = [&](v4f xv) {
    v4f o;
    o.x = xv.x * gc0 - xv.y * gs0;
    o.y = xv.x * gs0 + xv.y * gc0;
    o.z = xv.z * gc1 - xv.w * gs1;
    o.w = xv.z * gs1 + xv.w * gc1;
    return o;
  };

#if HAVE_ASYNC_LDS
  __shared__ v4f buf[2][TPB];
  if (blockFull) {
    // Prologue: stage tile 0 asynchronously into buffer 0.
    async_copy_b128(&buf[0][tid], x4 + blockBase + tid);

    for (int j = 0; j < TILES_PER_BLOCK; ++j) {
      const int tileStart = blockBase + j * TPB;   // multiple of 256

      if (j + 1 < TILES_PER_BLOCK) {
        // Stage next tile, then wait until only it is outstanding.
        async_copy_b128(&buf[(j + 1) & 1][tid], x4 + tileStart + TPB + tid);
        wait_async<1>();
      } else {
        wait_async<0>();
      }
      asm volatile("" ::: "memory");  // keep LDS read below the wait

      // Each 256-float4 tile lies entirely within one position.
      const int s = tileStart >> 9;   // /512 float4s per position
      if (s != cur_s) { cur_s = s; load_trig(s); }

      const v4f xv = buf[j & 1][tid];
      const v4f o  = rotate(xv);
      __builtin_nontemporal_store(o, &out4[tileStart + tid]);  // stream out
    }
    return;
  }
#endif

  // Tail / fallback path: direct (non-temporal) global loads.
  for (int j = 0; j < TILES_PER_BLOCK; ++j) {
    const int g = blockBase + j * TPB + tid;
    if (g >= n_f4) break;
    const int s = g >> 9;
    if (s != cur_s) { cur_s = s; load_trig(s); }
    const v4f xv = __builtin_nontemporal_load(&x4[g]);
    const v4f o  = rotate(xv);
    __builtin_nontemporal_store(o, &out4[g]);
  }
}

extern "C" void kernel_launch(void* const* d_in, const int* in_sizes, int n_in,
                              void* d_out, int out_size, void* d_ws, size_t ws_size,
                              hipStream_t stream) {
  (void)n_in; (void)out_size; (void)d_ws; (void)ws_size;

  const float* x   = (const float*)d_in[0];   // (1, s, 16, 128) f32
  const float* fc  = (const float*)d_in[1];   // (1024, 64) f32
  const float* fs  = (const float*)d_in[2];   // (1024, 64) f32
  /* d_in[3] = f (unused) */
  const int*   hp  = (const int*)d_in[4];
  const int*   wp  = (const int*)d_in[5];
  float*       out = (float*)d_out;

  const int n_f4   = in_sizes[0] / 4;                       // 16,773,120
  const int tiles  = (n_f4 + TPB - 1) / TPB;                // 65,520
  const int blocks = (tiles + TILES_PER_BLOCK - 1) / TILES_PER_BLOCK;  // 4,095

  rope3d_wan_s2v_kernel<<<blocks, TPB, 0, stream>>>(x, fc, fs, hp, wp, out, n_f4);
}